// acm_meta_path_hnode_prompt_layer_feature_weighted_sum_21534966022307
// MI455X (gfx1250) — compile-verified
//
#include <hip/hip_runtime.h>

// -----------------------------------------------------------------------------
// acm_meta_path feature-weighted gather/scatter (MI455X / gfx1250, wave32)
//
// out[n,d] = sum_{e : dst[e]==n} emb[src[e],d] * weight[d] * coef(e_feat[e])
// coef(f)  = sw0 + (f==0)*sw1 + (f==2)*sw2 + (f==4)*sw3 + (f==6)*sw4
//
// Memory-bound scatter/gather: no matmul structure -> WMMA not applicable.
// CDNA5 paths: global_load_async_to_lds_b128 (ASYNCcnt DMA of the weight row
// into LDS, amortized over a grid-stride loop), s_wait_asynccnt, NT-hinted
// streaming index loads to protect the L2 working set (emb + out both resident
// in the 192MB L2), coalesced B128 gathers, f32 atomics resolving in L2.
// -----------------------------------------------------------------------------

#define DIM 64
#define THREADS_PER_EDGE 16          // 16 lanes x float4 = 64 floats = one row
#define BLOCK_THREADS 256
#define EDGES_PER_BLOCK (BLOCK_THREADS / THREADS_PER_EDGE)
#define MAX_BLOCKS 4096              // grid-stride: ~15 batches/block at E=1M

__global__ void zero_out_kernel(float4* __restrict__ out4, int n4,
                                float* __restrict__ out, int n_total) {
    int i = blockIdx.x * blockDim.x + threadIdx.x;
    if (i < n4) out4[i] = make_float4(0.f, 0.f, 0.f, 0.f);
    int tail_base = n4 * 4;
    int t = tail_base + i;
    if (i < (n_total - tail_base)) out[t] = 0.f;
}

__global__ __launch_bounds__(BLOCK_THREADS)
void edge_scatter_kernel(const float* __restrict__ emb,
                         const int*   __restrict__ e_feat,
                         const int*   __restrict__ src_idx,
                         const int*   __restrict__ dst_idx,
                         const float* __restrict__ weight,
                         const float* __restrict__ sw,
                         float*       __restrict__ out,
                         int n_edges, int edges_per_sweep) {
    __shared__ __align__(16) float w_lds[DIM];

    // ---- Stage weight[0:64] (256B) into LDS via CDNA5 async DMA ----
    // 16 lanes of wave 0, 16 bytes each. Tracked by ASYNCcnt.
    if (threadIdx.x < (DIM * 4 / 16)) {
        unsigned lds_addr =
            (unsigned)(size_t)(&w_lds[0]) + threadIdx.x * 16u;   // low 32b of flat LDS addr == LDS offset
        unsigned long long gaddr =
            (unsigned long long)(size_t)weight + (unsigned long long)threadIdx.x * 16ull;
        asm volatile("global_load_async_to_lds_b128 %0, %1, off"
                     :: "v"(lds_addr), "v"(gaddr) : "memory");
    }
    asm volatile("s_wait_asynccnt 0" ::: "memory");   // waves with ASYNCcnt==0 pass immediately
    __syncthreads();

    const int chunk = (int)(threadIdx.x & 15);        // which float4 of the row

    // Loop-invariant: this thread's 4 weight values (one ds_load_b128, hoisted)
    const float4 w = *reinterpret_cast<const float4*>(&w_lds[chunk * 4]);

    // Uniform coefficients -> scalar loads, hoisted out of the loop
    const float sw0 = sw[0], sw1 = sw[1], sw2 = sw[2], sw3 = sw[3], sw4 = sw[4];

    for (int e = blockIdx.x * EDGES_PER_BLOCK + (int)(threadIdx.x >> 4);
         e < n_edges; e += edges_per_sweep) {
        // Streaming, read-once index data: NT hint keeps it out of the L2
        // working set (16 lanes hit the same dword -> one request per wave).
        const int ef = __builtin_nontemporal_load(e_feat + e);
        const int sn = __builtin_nontemporal_load(src_idx + e);
        const int dn = __builtin_nontemporal_load(dst_idx + e);

        float coef = sw0;
        coef += (ef == 0) ? sw1 : 0.0f;
        coef += (ef == 2) ? sw2 : 0.0f;
        coef += (ef == 4) ? sw3 : 0.0f;
        coef += (ef == 6) ? sw4 : 0.0f;

        // Coalesced 256B row gather: 16 lanes x global_load_b128 (RT policy:
        // avg degree ~10 -> strong L2 reuse on emb).
        const float4 x = *reinterpret_cast<const float4*>(
            emb + (size_t)sn * DIM + chunk * 4);

        // Scatter-add into out[dst] — resolves in L2 (out fits in 192MB L2)
        float* o = out + (size_t)dn * DIM + chunk * 4;
        atomicAdd(o + 0, x.x * w.x * coef);
        atomicAdd(o + 1, x.y * w.y * coef);
        atomicAdd(o + 2, x.z * w.z * coef);
        atomicAdd(o + 3, x.w * w.w * coef);
    }
}

extern "C" void kernel_launch(void* const* d_in, const int* in_sizes, int n_in,
                              void* d_out, int out_size, void* d_ws, size_t ws_size,
                              hipStream_t stream) {
    const float* emb    = (const float*)d_in[0];   // [N, 64] f32
    const int*   e_feat = (const int*)  d_in[1];   // [E] i32
    const int*   src    = (const int*)  d_in[2];   // [E] i32
    const int*   dst    = (const int*)  d_in[3];   // [E] i32
    const float* weight = (const float*)d_in[4];   // [1, 64] f32
    const float* sw     = (const float*)d_in[5];   // [1, 5] f32
    float* out = (float*)d_out;                    // [N, 64] f32

    const int n_edges = in_sizes[1];

    // Zero the accumulator first (harness poisons d_out; atomics accumulate).
    const int n4 = out_size / 4;
    {
        int zb = (n4 + BLOCK_THREADS - 1) / BLOCK_THREADS;
        if (zb < 1) zb = 1;
        zero_out_kernel<<<zb, BLOCK_THREADS, 0, stream>>>((float4*)out, n4, out, out_size);
    }

    // Grid-stride over edges: amortizes the LDS weight staging + barrier.
    int blocks = (n_edges + EDGES_PER_BLOCK - 1) / EDGES_PER_BLOCK;
    if (blocks > MAX_BLOCKS) blocks = MAX_BLOCKS;
    if (blocks < 1) blocks = 1;
    const int edges_per_sweep = blocks * EDGES_PER_BLOCK;

    edge_scatter_kernel<<<blocks, BLOCK_THREADS, 0, stream>>>(
        emb, e_feat, src, dst, weight, sw, out, n_edges, edges_per_sweep);
}